// DGCNN_63831803953204
// MI455X (gfx1250) — compile-verified
//
#include <hip/hip_runtime.h>
#include <hip/hip_bf16.h>

typedef __attribute__((ext_vector_type(2))) float v2f;
typedef __attribute__((ext_vector_type(8))) float v8f;

#define KNN 20
#define NEG_INF (-3.4e38f)

__device__ __forceinline__ v8f wmma_f32_4(v2f a, v2f b, v8f c) {
    // D = A(16x4) * B(4x16) + C(16x16), full fp32
    return __builtin_amdgcn_wmma_f32_16x16x4_f32(false, a, false, b, (short)0, c,
                                                 false, false);
}

__device__ __forceinline__ float lrelu(float v) { return v >= 0.f ? v : 0.2f * v; }
__device__ __forceinline__ float bnscale(float g) { return g * rsqrtf(1.0f + 1e-5f); }

// ---------------------------------------------------------------------------
// Gram matrix: G[b][m][n] = sum_c X[b][c][m] * X[b][c][n]
// grid: (N/16, (N/16)/8, B), block 256 (8 waves); one 16x16 tile per wave.
// Branch-free inner loop; K tail (layer 1, C=3) peeled with clamp+select.
// ---------------------------------------------------------------------------
__global__ void gram_kernel(const float* __restrict__ X, long bsX, int C, int N,
                            float* __restrict__ G) {
    const int lane = threadIdx.x & 31;
    const int wave = threadIdx.x >> 5;
    const int b  = blockIdx.z;
    const int N0 = blockIdx.x * 16;
    const int M0 = (blockIdx.y * 8 + wave) * 16;
    const float* Xb = X + (long)b * bsX;
    const int l16 = lane & 15;
    const int hi  = (lane >> 4) * 2;  // 0 or 2

    v8f acc = {0.f, 0.f, 0.f, 0.f, 0.f, 0.f, 0.f, 0.f};
    const int Cfull = C & ~3;
    const float* pa = Xb + hi * N + (M0 + l16);
    const float* pb = Xb + hi * N + (N0 + l16);
    for (int k = 0; k < Cfull; k += 4) {
        v2f a, bb;
        a.x  = pa[0];  a.y  = pa[N];
        bb.x = pb[0];  bb.y = pb[N];
        pa += 4 * N;   pb += 4 * N;
        acc = wmma_f32_4(a, bb, acc);
    }
    if (C & 3) {  // uniform branch; loads unconditional via clamped addresses
        const int r0 = Cfull + hi, r1 = r0 + 1;
        const int c0 = (r0 < C) ? r0 : (C - 1);
        const int c1 = (r1 < C) ? r1 : (C - 1);
        const float ax = Xb[c0 * N + M0 + l16];
        const float ay = Xb[c1 * N + M0 + l16];
        const float bx = Xb[c0 * N + N0 + l16];
        const float by = Xb[c1 * N + N0 + l16];
        v2f a, bb;
        a.x  = (r0 < C) ? ax : 0.f;  a.y  = (r1 < C) ? ay : 0.f;
        bb.x = (r0 < C) ? bx : 0.f;  bb.y = (r1 < C) ? by : 0.f;
        acc = wmma_f32_4(a, bb, acc);
    }
    float* Gb = G + (long)b * N * N;
    const int m0 = M0 + (lane >> 4) * 8;
#pragma unroll
    for (int r = 0; r < 8; ++r)
        Gb[(m0 + r) * N + N0 + l16] = acc[r];
}

// ---------------------------------------------------------------------------
// Top-k (k=20) of pd[m] = 2*G[n][m] - G[n][n] - G[m][m], one wave per row.
// ---------------------------------------------------------------------------
__global__ void topk_kernel(const float* __restrict__ G, int N, int* __restrict__ idx) {
    const int lane = threadIdx.x & 31;
    const int wave = threadIdx.x >> 5;
    const int row = blockIdx.x * 8 + wave;       // 0 .. B*N-1
    const int b = row >> 11;                     // N == 2048
    const int n = row & 2047;
    const float* Gb = G + (long)b * N * N;
    const float* Gr = Gb + n * N;
    const float dn = Gr[n];

    float vals[KNN];
    int   inds[KNN];
#pragma unroll
    for (int i = 0; i < KNN; ++i) { vals[i] = NEG_INF; inds[i] = 0; }

    for (int m = lane; m < N; m += 32) {
        float v = 2.f * Gr[m] - dn - Gb[m * N + m];
        if (v > vals[KNN - 1]) {
            int p = KNN - 1;
            while (p > 0 && vals[p - 1] < v) {
                vals[p] = vals[p - 1]; inds[p] = inds[p - 1]; --p;
            }
            vals[p] = v; inds[p] = m;
        }
    }
    // merge 32 sorted lists: 20 rounds of wave argmax on list heads
    int hp = 0;
    int* out = idx + row * KNN;
    for (int r = 0; r < KNN; ++r) {
        float mv = (hp < KNN) ? vals[hp] : NEG_INF;
        int   mi = (hp < KNN) ? inds[hp] : 0;
        int owner = lane;
        for (int off = 16; off; off >>= 1) {
            float ov = __shfl_xor(mv, off, 32);
            int   oi = __shfl_xor(mi, off, 32);
            int   oo = __shfl_xor(owner, off, 32);
            if (ov > mv || (ov == mv && oo < owner)) { mv = ov; mi = oi; owner = oo; }
        }
        if (lane == owner) ++hp;
        if (lane == 0) out[r] = mi;
    }
}

// ---------------------------------------------------------------------------
// Fused EdgeConv: gather edge feats -> LDS -> WMMA GEMM -> BN+LReLU -> max_k.
// One block (128 thr = 4 waves) per point (n, b).  K chunked through LDS.
// ---------------------------------------------------------------------------
template <int CIN, int OUT>
__global__ void edgeconv_kernel(const float* __restrict__ X, long bsX,
                                const float* __restrict__ W,
                                const float* __restrict__ gam,
                                const float* __restrict__ bet,
                                const int* __restrict__ idx,
                                float* __restrict__ Out, long bsO, int N) {
    constexpr int KDIM = 2 * CIN;
    constexpr int KPAD = (KDIM + 3) & ~3;
    constexpr int KC   = (KPAD < 64) ? KPAD : 64;   // LDS K-chunk
    constexpr int MTW  = (OUT / 16) / 4;            // M-tiles per wave

    __shared__ float E[KC][32];
    __shared__ int nbr[32];

    const int t    = threadIdx.x;          // 0..127
    const int lane = t & 31;
    const int wave = t >> 5;
    const int n = blockIdx.x;
    const int b = blockIdx.y;
    const float* Xb = X + (long)b * bsX;

    if (t < KNN) nbr[t] = idx[(b * N + n) * KNN + t];
    __syncthreads();

    const int l16 = lane & 15;
    const int hi  = (lane >> 4) * 2;

    v8f acc[MTW][2];
#pragma unroll
    for (int i = 0; i < MTW; ++i) {
        v8f z = {0.f, 0.f, 0.f, 0.f, 0.f, 0.f, 0.f, 0.f};
        acc[i][0] = z; acc[i][1] = z;
    }

    for (int kb = 0; kb < KPAD; kb += KC) {
        // build E chunk: rows [0,CIN) = nbr-ctr, [CIN,2CIN) = ctr, else 0
        for (int e = t; e < KC * 32; e += 128) {
            const int r = e >> 5, j = e & 31;
            const int kr = kb + r;
            float v = 0.f;
            if (j < KNN) {
                if (kr < CIN) {
                    v = Xb[kr * N + nbr[j]] - Xb[kr * N + n];
                } else if (kr < KDIM) {
                    v = Xb[(kr - CIN) * N + n];
                }
            }
            E[r][j] = v;
        }
        __syncthreads();

#pragma unroll
        for (int mi = 0; mi < MTW; ++mi) {
            const int mt = wave + mi * 4;
            const int orow = mt * 16 + l16;
            const float* wp = W + orow * KDIM + kb;
#pragma unroll
            for (int kk = 0; kk < KC; kk += 4) {
                const int kg = kk + hi;
                v2f a, b0, b1;
                if constexpr (KPAD == KDIM) {
                    a.x = wp[kg];
                    a.y = wp[kg + 1];
                } else {  // layer 1 only: clamp+select, no divergent branch
                    const int g0 = kb + kg, g1 = g0 + 1;
                    const int c0 = (g0 < KDIM) ? g0 : (KDIM - 1);
                    const int c1 = (g1 < KDIM) ? g1 : (KDIM - 1);
                    const float ax = W[orow * KDIM + c0];
                    const float ay = W[orow * KDIM + c1];
                    a.x = (g0 < KDIM) ? ax : 0.f;
                    a.y = (g1 < KDIM) ? ay : 0.f;
                }
                b0.x = E[kg][l16];       b0.y = E[kg + 1][l16];
                b1.x = E[kg][16 + l16];  b1.y = E[kg + 1][16 + l16];
                acc[mi][0] = wmma_f32_4(a, b0, acc[mi][0]);
                acc[mi][1] = wmma_f32_4(a, b1, acc[mi][1]);
            }
        }
        __syncthreads();
    }

    // BN + LeakyReLU + masked max over k (cols >= KNN masked to -inf)
    float* Ob = Out + (long)b * bsO;
#pragma unroll
    for (int mi = 0; mi < MTW; ++mi) {
        const int mt = wave + mi * 4;
        const int obase = mt * 16 + (lane >> 4) * 8;
#pragma unroll
        for (int r = 0; r < 8; ++r) {
            const int o = obase + r;
            const float sc = bnscale(gam[o]);
            const float bi = bet[o];
            float v0 = lrelu(acc[mi][0][r] * sc + bi);          // cols 0..15
            float v1 = lrelu(acc[mi][1][r] * sc + bi);          // cols 16..31
            if (16 + l16 >= KNN) v1 = NEG_INF;
            float v = fmaxf(v0, v1);
            for (int off = 8; off; off >>= 1)
                v = fmaxf(v, __shfl_xor(v, off, 16));
            if (l16 == 0) Ob[o * N + n] = v;
        }
    }
}

// ---------------------------------------------------------------------------
// conv5: x5[b][o][n] = lrelu(bn(W5 x cat)); M=1024, N=2048, K=512 per batch.
// ---------------------------------------------------------------------------
__global__ void conv5_kernel(const float* __restrict__ cat, const float* __restrict__ W,
                             const float* __restrict__ gam, const float* __restrict__ bet,
                             float* __restrict__ x5, int N) {
    const int lane = threadIdx.x & 31;
    const int wave = threadIdx.x >> 5;
    const int b  = blockIdx.z;
    const int N0 = blockIdx.x * 16;
    const int M0 = (blockIdx.y * 8 + wave) * 16;
    const float* Cb = cat + (long)b * 512 * N;
    const int l16 = lane & 15;
    const int hi  = (lane >> 4) * 2;

    v8f acc = {0.f, 0.f, 0.f, 0.f, 0.f, 0.f, 0.f, 0.f};
    const float* wp = W + (M0 + l16) * 512 + hi;
    const float* pb = Cb + hi * N + (N0 + l16);
    for (int k = 0; k < 512; k += 4) {
        v2f a, bb;
        a.x  = wp[0];  a.y  = wp[1];
        bb.x = pb[0];  bb.y = pb[N];
        wp += 4;       pb += 4 * N;
        acc = wmma_f32_4(a, bb, acc);
    }
    float* Xb = x5 + (long)b * 1024 * N;
    const int m0 = M0 + (lane >> 4) * 8;
#pragma unroll
    for (int r = 0; r < 8; ++r) {
        const int o = m0 + r;
        float v = lrelu(acc[r] * bnscale(gam[o]) + bet[o]);
        Xb[o * N + N0 + l16] = v;
    }
}

// ---------------------------------------------------------------------------
// Global max + mean pooling over N; one wave per (b, o).
// ---------------------------------------------------------------------------
__global__ void pool_kernel(const float* __restrict__ x5, float* __restrict__ pooled,
                            float* __restrict__ outmap, int N) {
    const int lane = threadIdx.x & 31;
    const int wave = threadIdx.x >> 5;
    const int gid = blockIdx.x * 8 + wave;     // 0..8191
    const int b = gid >> 10, o = gid & 1023;
    const float* row = x5 + ((long)b * 1024 + o) * N;
    float mx = NEG_INF, sm = 0.f;
    for (int i = lane; i < N; i += 32) { float v = row[i]; mx = fmaxf(mx, v); sm += v; }
    for (int off = 16; off; off >>= 1) {
        mx = fmaxf(mx, __shfl_xor(mx, off, 32));
        sm += __shfl_xor(sm, off, 32);
    }
    if (lane == 0) {
        const float mean = sm / (float)N;
        pooled[b * 2048 + o] = mx;
        pooled[b * 2048 + 1024 + o] = mean;
        outmap[b * 2048 + o] = mx;
        outmap[b * 2048 + 1024 + o] = mean;
    }
}

// ---------------------------------------------------------------------------
// Pose branch: p1 = lrelu(bn(posefeat @ Wp^T + bp))   (8 x 32)
// ---------------------------------------------------------------------------
__global__ void pose_kernel(const float* __restrict__ pf, const float* __restrict__ Wp,
                            const float* __restrict__ bp, const float* __restrict__ gp,
                            const float* __restrict__ bpb, float* __restrict__ p1) {
    const int t = threadIdx.x;
    if (t < 256) {
        const int b = t >> 5, j = t & 31;
        float s = bp[j];
        for (int c = 0; c < 6; ++c) s += pf[b * 6 + c] * Wp[j * 6 + c];
        s = s * bnscale(gp[j]) + bpb[j];
        p1[b * 32 + j] = lrelu(s);
    }
}

// ---------------------------------------------------------------------------
// Dense layer: out[b][j] = act(bn(dot([in1|in2], W[j]) + bv[j])); in2 optional.
// ---------------------------------------------------------------------------
__global__ void fc_kernel(const float* __restrict__ in1, int K1,
                          const float* __restrict__ in2, int K2,
                          const float* __restrict__ Wm, const float* __restrict__ bv,
                          const float* __restrict__ gam, const float* __restrict__ bet,
                          float* __restrict__ out, int O, int act) {
    const int b = blockIdx.x;
    for (int j = threadIdx.x; j < O; j += blockDim.x) {
        const float* w = Wm + j * (K1 + K2);
        float s = bv[j];
        const float* x1 = in1 + b * K1;
        for (int c = 0; c < K1; ++c) s += x1[c] * w[c];
        if (in2) {
            const float* x2 = in2 + b * K2;
            for (int c = 0; c < K2; ++c) s += x2[c] * w[K1 + c];
        }
        if (gam) s = s * bnscale(gam[j]) + bet[j];
        if (act) s = lrelu(s);
        out[b * O + j] = s;
    }
}

// ---------------------------------------------------------------------------
extern "C" void kernel_launch(void* const* d_in, const int* in_sizes, int n_in,
                              void* d_out, int out_size, void* d_ws, size_t ws_size,
                              hipStream_t stream) {
    (void)in_sizes; (void)n_in; (void)out_size; (void)ws_size;
    const int B = 8, N = 2048;

    const float* x   = (const float*)d_in[0];
    const float* pf  = (const float*)d_in[1];
    const float* W1  = (const float*)d_in[2];
    const float* g1  = (const float*)d_in[3];
    const float* b1  = (const float*)d_in[4];
    const float* W2  = (const float*)d_in[5];
    const float* g2  = (const float*)d_in[6];
    const float* b2  = (const float*)d_in[7];
    const float* W3  = (const float*)d_in[8];
    const float* g3  = (const float*)d_in[9];
    const float* b3  = (const float*)d_in[10];
    const float* W4  = (const float*)d_in[11];
    const float* g4  = (const float*)d_in[12];
    const float* b4  = (const float*)d_in[13];
    const float* W5  = (const float*)d_in[14];
    const float* g5  = (const float*)d_in[15];
    const float* b5  = (const float*)d_in[16];
    const float* Wp  = (const float*)d_in[17];
    const float* bp  = (const float*)d_in[18];
    const float* gp  = (const float*)d_in[19];
    const float* bpb = (const float*)d_in[20];
    const float* WL1 = (const float*)d_in[21];
    const float* bL1 = (const float*)d_in[22];
    const float* g6  = (const float*)d_in[23];
    const float* b6  = (const float*)d_in[24];
    const float* WL2 = (const float*)d_in[25];
    const float* bL2 = (const float*)d_in[26];
    const float* g7  = (const float*)d_in[27];
    const float* b7  = (const float*)d_in[28];
    const float* WL3 = (const float*)d_in[29];
    const float* bL3 = (const float*)d_in[30];

    float* logits = (float*)d_out;           // (8, 40)
    float* outmap = logits + B * 40;         // (8, 2048)

    // workspace carve-up (floats)
    float* ws     = (float*)d_ws;
    float* gram   = ws;                                       // B*N*N
    float* cat    = gram + (size_t)B * N * N;                 // B*512*N
    int*   nidx   = (int*)(cat + (size_t)B * 512 * N);        // B*N*KNN
    float* pooled = (float*)(nidx + (size_t)B * N * KNN);     // B*2048
    float* p1     = pooled + (size_t)B * 2048;                // B*32
    float* h1     = p1 + (size_t)B * 32;                      // B*512
    float* h2     = h1 + (size_t)B * 512;                     // B*256
    float* x5     = gram;   // reuse: gram is dead after layer 4

    const dim3 gGrid(N / 16, (N / 16) / 8, B);
    const dim3 tGrid((B * N) / 8);
    const dim3 eGrid(N, B);
    const long bsCat = (long)512 * N;

    // ---- layer 1 (in: x, 3ch) -> cat[:, 0:64] ----
    gram_kernel<<<gGrid, 256, 0, stream>>>(x, (long)3 * N, 3, N, gram);
    topk_kernel<<<tGrid, 256, 0, stream>>>(gram, N, nidx);
    edgeconv_kernel<3, 64><<<eGrid, 128, 0, stream>>>(x, (long)3 * N, W1, g1, b1,
                                                      nidx, cat, bsCat, N);
    // ---- layer 2 (in: x1, 64ch) -> cat[:, 64:128] ----
    gram_kernel<<<gGrid, 256, 0, stream>>>(cat, bsCat, 64, N, gram);
    topk_kernel<<<tGrid, 256, 0, stream>>>(gram, N, nidx);
    edgeconv_kernel<64, 64><<<eGrid, 128, 0, stream>>>(cat, bsCat, W2, g2, b2,
                                                       nidx, cat + (size_t)64 * N, bsCat, N);
    // ---- layer 3 (in: x2, 64ch) -> cat[:, 128:256] ----
    gram_kernel<<<gGrid, 256, 0, stream>>>(cat + (size_t)64 * N, bsCat, 64, N, gram);
    topk_kernel<<<tGrid, 256, 0, stream>>>(gram, N, nidx);
    edgeconv_kernel<64, 128><<<eGrid, 128, 0, stream>>>(cat + (size_t)64 * N, bsCat,
                                                        W3, g3, b3, nidx,
                                                        cat + (size_t)128 * N, bsCat, N);
    // ---- layer 4 (in: x3, 128ch) -> cat[:, 256:512] ----
    gram_kernel<<<gGrid, 256, 0, stream>>>(cat + (size_t)128 * N, bsCat, 128, N, gram);
    topk_kernel<<<tGrid, 256, 0, stream>>>(gram, N, nidx);
    edgeconv_kernel<128, 256><<<eGrid, 128, 0, stream>>>(cat + (size_t)128 * N, bsCat,
                                                         W4, g4, b4, nidx,
                                                         cat + (size_t)256 * N, bsCat, N);
    // ---- conv5: (1024 x 512) x (512 x N) per batch ----
    conv5_kernel<<<dim3(N / 16, (1024 / 16) / 8, B), 256, 0, stream>>>(cat, W5, g5, b5, x5, N);
    // ---- pooling -> pooled / map_ output ----
    pool_kernel<<<dim3((B * 1024) / 8), 256, 0, stream>>>(x5, pooled, outmap, N);
    // ---- head ----
    pose_kernel<<<1, 256, 0, stream>>>(pf, Wp, bp, gp, bpb, p1);
    fc_kernel<<<dim3(B), 512, 0, stream>>>(pooled, 2048, p1, 32, WL1, bL1, g6, b6, h1, 512, 1);
    fc_kernel<<<dim3(B), 256, 0, stream>>>(h1, 512, nullptr, 0, WL2, bL2, g7, b7, h2, 256, 1);
    fc_kernel<<<dim3(B), 64, 0, stream>>>(h2, 256, nullptr, 0, WL3, bL3, nullptr, nullptr,
                                          logits, 40, 0);
}